// LGCN_Encoder_38276748541965
// MI455X (gfx1250) — compile-verified
//
#include <hip/hip_runtime.h>
#include <hip/hip_bf16.h>

#define EMB 64
#define TILE 256          // edges staged per block tile
#define SPMM_THREADS 256  // 8 wave32s; 16 lanes per edge -> 16 edges per pass

// ---------------------------------------------------------------------------
// Helpers
// ---------------------------------------------------------------------------
__device__ __forceinline__ unsigned lds_off_u32(const void* p) {
    // __shared__ generic pointers carry the LDS byte offset in the low 32 bits
    return (unsigned)(unsigned long long)p;
}

__device__ __forceinline__ void async_load_b32_to_lds(unsigned lds_addr,
                                                      unsigned byte_off,
                                                      unsigned long long sbase) {
    // gfx1250 async global->LDS DMA (per-lane addresses), tracked by ASYNCcnt
    asm volatile("global_load_async_to_lds_b32 %0, %1, %2"
                 :
                 : "v"(lds_addr), "v"(byte_off), "s"(sbase)
                 : "memory");
}

__device__ __forceinline__ void wait_asynccnt0() {
    asm volatile("s_wait_asynccnt 0" ::: "memory");
}

// Native no-return f32 atomic add (L2-resolved), SADDR form:
// base in SGPR pair, per-lane signed 32-bit byte offset in a VGPR,
// consecutive dwords addressed via the immediate offset field.
__device__ __forceinline__ void atomic_add4_f32(unsigned long long ybase,
                                                unsigned byte_off,
                                                float v0, float v1,
                                                float v2, float v3) {
    asm volatile("global_atomic_add_f32 %0, %1, %2"
                 :: "v"(byte_off), "v"(v0), "s"(ybase) : "memory");
    asm volatile("global_atomic_add_f32 %0, %1, %2 offset:4"
                 :: "v"(byte_off), "v"(v1), "s"(ybase) : "memory");
    asm volatile("global_atomic_add_f32 %0, %1, %2 offset:8"
                 :: "v"(byte_off), "v"(v2), "s"(ybase) : "memory");
    asm volatile("global_atomic_add_f32 %0, %1, %2 offset:12"
                 :: "v"(byte_off), "v"(v3), "s"(ybase) : "memory");
}

// ---------------------------------------------------------------------------
// ego = concat(user,item); h0 = ego; acc(d_out) = ego; h1 = 0
// ---------------------------------------------------------------------------
__global__ void lgcn_init_kernel(const float* __restrict__ user,
                                 const float* __restrict__ item,
                                 float* __restrict__ h0,
                                 float* __restrict__ h1,
                                 float* __restrict__ acc,
                                 int userElems, int totalElems) {
    const int n4 = totalElems >> 2;
    const int u4 = userElems >> 2;
    const int stride = gridDim.x * blockDim.x;
    const float4* uf = (const float4*)user;
    const float4* itf = (const float4*)item;
    float4* h04 = (float4*)h0;
    float4* h14 = (float4*)h1;
    float4* ac4 = (float4*)acc;
    const float4 z = make_float4(0.f, 0.f, 0.f, 0.f);
    for (int i = blockIdx.x * blockDim.x + threadIdx.x; i < n4; i += stride) {
        float4 v = (i < u4) ? uf[i] : itf[i - u4];
        h04[i] = v;
        ac4[i] = v;
        h14[i] = z;
    }
}

// ---------------------------------------------------------------------------
// acc += h; z = 0   (fused layer-accumulate + clear of next output buffer)
// ---------------------------------------------------------------------------
__global__ void lgcn_add_zero_kernel(float* __restrict__ acc,
                                     const float* __restrict__ h,
                                     float* __restrict__ z,
                                     int totalElems) {
    const int n4 = totalElems >> 2;
    const int stride = gridDim.x * blockDim.x;
    const float4* h4 = (const float4*)h;
    float4* a4 = (float4*)acc;
    float4* z4 = (float4*)z;
    const float4 zero = make_float4(0.f, 0.f, 0.f, 0.f);
    for (int i = blockIdx.x * blockDim.x + threadIdx.x; i < n4; i += stride) {
        float4 a = a4[i];
        float4 b = h4[i];
        a.x += b.x; a.y += b.y; a.z += b.z; a.w += b.w;
        a4[i] = a;
        z4[i] = zero;
    }
}

// ---------------------------------------------------------------------------
// acc = (acc + h) * 0.25   (final layer accumulate fused with the /4 mean)
// ---------------------------------------------------------------------------
__global__ void lgcn_final_kernel(float* __restrict__ acc,
                                  const float* __restrict__ h,
                                  int totalElems) {
    const int n4 = totalElems >> 2;
    const int stride = gridDim.x * blockDim.x;
    const float4* h4 = (const float4*)h;
    float4* a4 = (float4*)acc;
    for (int i = blockIdx.x * blockDim.x + threadIdx.x; i < n4; i += stride) {
        float4 a = a4[i];
        float4 b = h4[i];
        a.x = (a.x + b.x) * 0.25f;
        a.y = (a.y + b.y) * 0.25f;
        a.z = (a.z + b.z) * 0.25f;
        a.w = (a.w + b.w) * 0.25f;
        a4[i] = a;
    }
}

// ---------------------------------------------------------------------------
// COO SpMM: y[row[e], :] += val[e] * x[col[e], :]   (y pre-zeroed)
// COO streams (row/col/val) are double-buffered into LDS via async DMA.
// 16 lanes per edge, float4 (16B) per lane -> together the full 256B row.
// Gathers hit the L2-resident embedding table; scatters are native
// global_atomic_add_f32 (no-return) resolved in L2.
// ---------------------------------------------------------------------------
__global__ __launch_bounds__(SPMM_THREADS)
void lgcn_spmm_kernel(const int* __restrict__ rows,
                      const int* __restrict__ cols,
                      const float* __restrict__ vals,
                      const float* __restrict__ x,
                      float* __restrict__ y,
                      int nnz) {
    __shared__ int s_row[2][TILE];
    __shared__ int s_col[2][TILE];
    __shared__ float s_val[2][TILE];

    const int tid = threadIdx.x;
    const int numTiles = (nnz + TILE - 1) / TILE;
    int tile = blockIdx.x;
    if (tile >= numTiles) return;

    const unsigned long long rB = (unsigned long long)rows;
    const unsigned long long cB = (unsigned long long)cols;
    const unsigned long long vB = (unsigned long long)vals;
    const unsigned long long yB = (unsigned long long)y;

    int buf = 0;
    // Stage first tile asynchronously (one b32 per lane per stream).
    {
        int e = tile * TILE + tid;
        if (e >= nnz) e = nnz - 1;
        const unsigned off = (unsigned)e * 4u;
        async_load_b32_to_lds(lds_off_u32(&s_row[buf][tid]), off, rB);
        async_load_b32_to_lds(lds_off_u32(&s_col[buf][tid]), off, cB);
        async_load_b32_to_lds(lds_off_u32(&s_val[buf][tid]), off, vB);
    }

    const int g = tid >> 4;               // edge slot within a pass (0..15)
    const int lane4 = (tid & 15) * 4;     // float offset within the 64-wide row

    for (; tile < numTiles; tile += gridDim.x) {
        wait_asynccnt0();                 // this wave's DMA landed in LDS
        __syncthreads();                  // all waves' DMA visible

        // Prefetch the next tile into the other buffer while we compute.
        const int nextTile = tile + (int)gridDim.x;
        if (nextTile < numTiles) {
            const int nb = buf ^ 1;
            int e = nextTile * TILE + tid;
            if (e >= nnz) e = nnz - 1;
            const unsigned off = (unsigned)e * 4u;
            async_load_b32_to_lds(lds_off_u32(&s_row[nb][tid]), off, rB);
            async_load_b32_to_lds(lds_off_u32(&s_col[nb][tid]), off, cB);
            async_load_b32_to_lds(lds_off_u32(&s_val[nb][tid]), off, vB);
        }

        const int base = tile * TILE;
#pragma unroll 4
        for (int pass = 0; pass < TILE / 16; ++pass) {
            const int el = pass * 16 + g;
            const int e = base + el;
            if (e < nnz) {
                const int c = s_col[buf][el];
                const int r = s_row[buf][el];
                const float v = s_val[buf][el];
                const float4 xv =
                    *(const float4*)(x + (size_t)c * EMB + lane4);
                const unsigned yoff = ((unsigned)r * EMB + (unsigned)lane4) * 4u;
                atomic_add4_f32(yB, yoff, v * xv.x, v * xv.y, v * xv.z, v * xv.w);
            }
        }
        __syncthreads();                  // everyone done reading buf
        buf ^= 1;
    }
}

// ---------------------------------------------------------------------------
// Host launcher
// ---------------------------------------------------------------------------
extern "C" void kernel_launch(void* const* d_in, const int* in_sizes, int n_in,
                              void* d_out, int out_size, void* d_ws, size_t ws_size,
                              hipStream_t stream) {
    const float* user = (const float*)d_in[0];
    const float* item = (const float*)d_in[1];
    const int* adj_row = (const int*)d_in[2];
    const int* adj_col = (const int*)d_in[3];
    const float* adj_vals = (const float*)d_in[4];

    const int userElems = in_sizes[0];            // USER_NUM * 64
    const int itemElems = in_sizes[1];            // ITEM_NUM * 64
    const int nnz = in_sizes[2];
    const int totalElems = userElems + itemElems; // N * 64

    float* acc = (float*)d_out;                   // running sum lives in d_out
    float* h0 = (float*)d_ws;
    float* h1 = h0 + (size_t)totalElems;

    const int denseBlocks = 4096;
    const int denseThreads = 256;

    const int numTiles = (nnz + TILE - 1) / TILE;
    int spmmBlocks = numTiles < 3072 ? numTiles : 3072;

    // ego -> h0, acc; h1 zeroed (output of layer 1)
    lgcn_init_kernel<<<denseBlocks, denseThreads, 0, stream>>>(
        user, item, h0, h1, acc, userElems, totalElems);

    // layer 1: h1 = A * h0 ; acc += h1 ; zero h0
    lgcn_spmm_kernel<<<spmmBlocks, SPMM_THREADS, 0, stream>>>(
        adj_row, adj_col, adj_vals, h0, h1, nnz);
    lgcn_add_zero_kernel<<<denseBlocks, denseThreads, 0, stream>>>(
        acc, h1, h0, totalElems);

    // layer 2: h0 = A * h1 ; acc += h0 ; zero h1
    lgcn_spmm_kernel<<<spmmBlocks, SPMM_THREADS, 0, stream>>>(
        adj_row, adj_col, adj_vals, h1, h0, nnz);
    lgcn_add_zero_kernel<<<denseBlocks, denseThreads, 0, stream>>>(
        acc, h0, h1, totalElems);

    // layer 3: h1 = A * h0 ; acc = (acc + h1) / 4
    lgcn_spmm_kernel<<<spmmBlocks, SPMM_THREADS, 0, stream>>>(
        adj_row, adj_col, adj_vals, h0, h1, nnz);
    lgcn_final_kernel<<<denseBlocks, denseThreads, 0, stream>>>(
        acc, h1, totalElems);
}